// UpsampleLoss_9560597200959
// MI455X (gfx1250) — compile-verified
//
#include <hip/hip_runtime.h>

// ---------------------------------------------------------------------------
// UpsampleLoss on MI455X (gfx1250): fused WMMA pairwise-distance kernels.
// Never materializes the 8x4096x4096 matrices (saves ~1 GB of HBM traffic).
// Data staging uses CDNA5-specific paths:
//   kernel A: Tensor Data Mover (tensor_load_to_lds + s_wait_tensorcnt)
//   kernel B: async global->LDS copies (global_load_async_to_lds_b128,
//             s_wait_asynccnt), double-buffered to overlap DMA with WMMA.
// ---------------------------------------------------------------------------

typedef float    v2f __attribute__((ext_vector_type(2)));
typedef float    v8f __attribute__((ext_vector_type(8)));
typedef unsigned v4u __attribute__((ext_vector_type(4)));
typedef unsigned v8u __attribute__((ext_vector_type(8)));

#define NPTS   4096
#define NB     8
#define CHUNK  1024          // points staged in LDS per pass (12 KB)
#define RADIUS_C 0.07f
#define EPS_C    1e-12f

// Build a 16x4 (K padded 3->4) WMMA fragment for point `base+l16`.
// 32-bit A/B layout: lanes 0-15 hold K=0,1 in v0,v1; lanes 16-31 hold K=2,3.
__device__ __forceinline__ v2f frag_from_xyz(float x, float y, float z, int lane) {
    v2f f;
    f.x = (lane < 16) ? x : z;
    f.y = (lane < 16) ? y : 0.0f;
    return f;
}

// ---------------------------------------------------------------------------
// TDM: DMA `n` contiguous f32 elements from global -> LDS (1-D tile).
// D# per CDNA5 ISA ch.8: group0 = {count=1, lds_addr, global_addr[56:0],
// type=2}; group1 = {data_size=4B, tensor_dim0=n, tensor_dim1=1, tile_dim0=n,
// tensor_dim0_stride=n}. Issue once per wave; EXEC is ignored by TDM.
// ---------------------------------------------------------------------------
__device__ __forceinline__ void tdm_load_f32(unsigned lds_off,
                                             unsigned long long gaddr,
                                             unsigned n /* < 65536 */) {
    v4u g0;
    g0[0] = 1u;                                        // count=1, user mode
    g0[1] = lds_off;                                   // lds_addr (bytes)
    g0[2] = (unsigned)(gaddr & 0xFFFFFFFFull);         // global_addr[31:0]
    g0[3] = (unsigned)((gaddr >> 32) & 0x01FFFFFFull)  // global_addr[56:32]
          | (2u << 30);                                // type = 2 ("image")
    v8u g1;
    g1[0] = (2u << 16);                                // data_size = 4 bytes
    g1[1] = (n & 0xFFFFu) << 16;                       // tensor_dim0[15:0]
    g1[2] = (n >> 16) | (1u << 16);                    // tensor_dim0[31:16], tensor_dim1=1
    g1[3] = (n & 0xFFFFu) << 16;                       // tile_dim0 = n
    g1[4] = 0u;                                        // tile_dim1/2 unused
    g1[5] = n;                                         // tensor_dim0_stride[31:0]
    g1[6] = 0u;
    g1[7] = 0u;
    asm volatile("tensor_load_to_lds %0, %1" :: "s"(g0), "s"(g1) : "memory");
}

// Per-lane async 16-byte copy global -> LDS (tracked by ASYNCcnt).
__device__ __forceinline__ void async_copy_b128(const float* gsrc, float* ldst) {
    unsigned loff = (unsigned)(uintptr_t)ldst;              // LDS byte offset
    unsigned long long ga = (unsigned long long)(uintptr_t)gsrc;
    asm volatile("global_load_async_to_lds_b128 %0, %1, off"
                 :: "v"(loff), "v"(ga) : "memory");
}

__device__ __forceinline__ void wait_asynccnt0() {
    asm volatile("s_wait_asynccnt 0x0" ::: "memory");
}

// ---------------------------------------------------------------------------
// Kernel A: per pred point, min squared distance to any gt point (EMD term).
// Grid: (NPTS/128, NB); block 256 = 8 waves; each wave owns 16 pred rows.
// gt chunks staged into LDS by the Tensor Data Mover.
// ---------------------------------------------------------------------------
__global__ __launch_bounds__(256) void emd_min_kernel(
    const float* __restrict__ pred, const float* __restrict__ gt,
    float* __restrict__ ws_min)
{
    __shared__ __align__(16) float sp[CHUNK * 3];
    const int b = blockIdx.y;
    const float* prb = pred + (size_t)b * NPTS * 3;
    const float* gtb = gt   + (size_t)b * NPTS * 3;

    const int wave = threadIdx.x >> 5;
    const int lane = threadIdx.x & 31;
    const int l16  = lane & 15;
    const int hf   = lane >> 4;
    const int rowbase = blockIdx.x * 128 + wave * 16;

    // A fragment (pred rows), pre-scaled by -2 so WMMA yields -2*(p.g)
    const int rA = rowbase + l16;
    v2f a = frag_from_xyz(prb[3*rA], prb[3*rA+1], prb[3*rA+2], lane);
    a.x *= -2.0f; a.y *= -2.0f;

    float aa8[8];
    #pragma unroll
    for (int v = 0; v < 8; ++v) {
        int r = rowbase + v + 8*hf;
        float x = prb[3*r], y = prb[3*r+1], z = prb[3*r+2];
        aa8[v] = x*x + y*y + z*z;
    }

    float minv[8];
    #pragma unroll
    for (int v = 0; v < 8; ++v) minv[v] = 3.4e38f;

    for (int c0 = 0; c0 < NPTS; c0 += CHUNK) {
        __syncthreads();                       // all waves done with prev chunk
        if (wave == 0) {                       // one TDM DMA per chunk
            tdm_load_f32((unsigned)(uintptr_t)&sp[0],
                         (unsigned long long)(uintptr_t)(gtb + (size_t)c0 * 3),
                         CHUNK * 3);
            __builtin_amdgcn_s_wait_tensorcnt(0);
        }
        __syncthreads();                       // chunk visible to all waves
        for (int t = 0; t < CHUNK/16; ++t) {
            const int pl = t*16 + l16;
            float x = sp[3*pl], y = sp[3*pl+1], z = sp[3*pl+2];
            float bb = x*x + y*y + z*z;
            v2f bf = frag_from_xyz(x, y, z, lane);
            v8f c = {};
            c = __builtin_amdgcn_wmma_f32_16x16x4_f32(
                    false, a, false, bf, (short)0, c, false, false);
            #pragma unroll
            for (int v = 0; v < 8; ++v) {
                float d2 = aa8[v] + bb + c[v];   // |p|^2 + |g|^2 - 2 p.g
                minv[v] = fminf(minv[v], d2);
            }
        }
    }
    // min across the 16 lanes that share each row (two independent halves)
    #pragma unroll
    for (int m = 8; m >= 1; m >>= 1) {
        #pragma unroll
        for (int v = 0; v < 8; ++v)
            minv[v] = fminf(minv[v], __shfl_xor(minv[v], m, 16));
    }
    if (l16 == 0) {
        #pragma unroll
        for (int v = 0; v < 8; ++v)
            ws_min[(size_t)b * NPTS + rowbase + v + 8*hf] = minv[v];
    }
}

// ---------------------------------------------------------------------------
// Kernel B: 4 nearest pred-pred neighbors (self excluded) -> repulsion term.
// pred chunks staged via double-buffered async global->LDS copies so the DMA
// of chunk c+1 overlaps the WMMA + 4-NN insertion work on chunk c.
// ---------------------------------------------------------------------------
__global__ __launch_bounds__(256) void rep_knn_kernel(
    const float* __restrict__ pred, float* __restrict__ ws_rep)
{
    __shared__ __align__(16) float spb[2][CHUNK * 3];   // 24 KB double buffer
    __shared__ float sc[8][16][16][4];                  // 32 KB merge scratch
    const int b = blockIdx.y;
    const float* prb = pred + (size_t)b * NPTS * 3;

    const int wave = threadIdx.x >> 5;
    const int lane = threadIdx.x & 31;
    const int l16  = lane & 15;
    const int hf   = lane >> 4;
    const int rowbase = blockIdx.x * 128 + wave * 16;

    const int rA = rowbase + l16;
    v2f a = frag_from_xyz(prb[3*rA], prb[3*rA+1], prb[3*rA+2], lane);
    a.x *= -2.0f; a.y *= -2.0f;

    float aa8[8];
    #pragma unroll
    for (int v = 0; v < 8; ++v) {
        int r = rowbase + v + 8*hf;
        float x = prb[3*r], y = prb[3*r+1], z = prb[3*r+2];
        aa8[v] = x*x + y*y + z*z;
    }

    float best[8][4];
    #pragma unroll
    for (int v = 0; v < 8; ++v) {
        #pragma unroll
        for (int j = 0; j < 4; ++j) best[v][j] = 3.4e38f;
    }

    // preload chunk 0 into buffer 0 (3 x 16B per thread = 12 KB / block)
    #pragma unroll
    for (int s = 0; s < 3; ++s) {
        const int fi = s * 1024 + threadIdx.x * 4;
        async_copy_b128(prb + fi, &spb[0][fi]);
    }
    wait_asynccnt0();
    __syncthreads();

    const int nch = NPTS / CHUNK;
    for (int ci = 0; ci < nch; ++ci) {
        const int c0 = ci * CHUNK;
        const int cb = ci & 1;
        if (ci + 1 < nch) {                    // prefetch next chunk
            const float* gnext = prb + (size_t)(c0 + CHUNK) * 3;
            #pragma unroll
            for (int s = 0; s < 3; ++s) {
                const int fi = s * 1024 + threadIdx.x * 4;
                async_copy_b128(gnext + fi, &spb[cb ^ 1][fi]);
            }
        }
        const float* sp = &spb[cb][0];
        for (int t = 0; t < CHUNK/16; ++t) {
            const int pl  = t*16 + l16;
            const int col = c0 + pl;
            float x = sp[3*pl], y = sp[3*pl+1], z = sp[3*pl+2];
            float bb = x*x + y*y + z*z;
            v2f bf = frag_from_xyz(x, y, z, lane);
            v8f c = {};
            c = __builtin_amdgcn_wmma_f32_16x16x4_f32(
                    false, a, false, bf, (short)0, c, false, false);
            #pragma unroll
            for (int v = 0; v < 8; ++v) {
                const int row = rowbase + v + 8*hf;
                float d2 = aa8[v] + bb + c[v];
                if ((col != row) && (d2 < best[v][3])) {
                    float b0 = best[v][0], b1 = best[v][1], b2 = best[v][2];
                    if (d2 < b2) {
                        best[v][3] = b2;
                        if (d2 < b1) {
                            best[v][2] = b1;
                            if (d2 < b0) { best[v][1] = b0; best[v][0] = d2; }
                            else          { best[v][1] = d2; }
                        } else { best[v][2] = d2; }
                    } else { best[v][3] = d2; }
                }
            }
        }
        wait_asynccnt0();                      // next buffer's DMA complete
        __syncthreads();                       // all waves done reading spb[cb]
    }

    // merge 16 per-lane candidate lists per row via LDS
    #pragma unroll
    for (int v = 0; v < 8; ++v) {
        const int row = v + 8*hf;
        #pragma unroll
        for (int j = 0; j < 4; ++j) sc[wave][row][l16][j] = best[v][j];
    }
    __syncthreads();
    if (lane < 16) {
        const int row = lane;
        float m0 = 3.4e38f, m1 = m0, m2 = m0, m3 = m0;
        for (int l = 0; l < 16; ++l) {
            #pragma unroll
            for (int j = 0; j < 4; ++j) {
                float d2 = sc[wave][row][l][j];
                if (d2 < m3) {
                    if (d2 < m2) {
                        m3 = m2;
                        if (d2 < m1) {
                            m2 = m1;
                            if (d2 < m0) { m1 = m0; m0 = d2; } else { m1 = d2; }
                        } else { m2 = d2; }
                    } else { m3 = d2; }
                }
            }
        }
        const float invH2 = 1.0f / (0.03f * 0.03f);
        float q[4] = {m0, m1, m2, m3};
        float s = 0.0f;
        #pragma unroll
        for (int j = 0; j < 4; ++j) {
            float r2   = fmaxf(q[j], EPS_C);
            float dist = sqrtf(r2);
            s += (RADIUS_C - dist) * expf(-r2 * invH2);
        }
        ws_rep[(size_t)b * NPTS + rowbase + row] = s;
    }
}

// ---------------------------------------------------------------------------
// Kernel C: TDA term. Per batch: subsample 16 points (deterministic hash in
// lieu of jax threefry permutation), Prim MST deaths, sort, L2 (Q=2).
// ---------------------------------------------------------------------------
__device__ __forceinline__ unsigned hash_u(unsigned x) {
    x ^= x >> 16; x *= 0x7FEB352Du;
    x ^= x >> 15; x *= 0x846CA68Bu;
    x ^= x >> 16;
    return x;
}

__global__ __launch_bounds__(32) void tda_kernel(
    const float* __restrict__ pred, const float* __restrict__ gt,
    float* __restrict__ ws_wd)
{
    __shared__ float pts[NB][16][3];
    __shared__ float dmat[NB][16][16];
    __shared__ float deaths[NB][2][15];
    const int b = threadIdx.x;
    if (b >= NB) return;
    for (int pass = 0; pass < 2; ++pass) {
        const float* src = (pass == 0) ? pred : gt;
        const unsigned salt = (pass == 0) ? 0x12345678u : 0x9E3779B9u;
        for (int j = 0; j < 16; ++j) {
            unsigned h = hash_u((unsigned)j * 0x9E3779B1u +
                                (unsigned)b * 0x85EBCA77u + salt);
            int ix = (int)(h & (NPTS - 1));
            for (int d = 0; d < 3; ++d)
                pts[b][j][d] = src[((size_t)b * NPTS + ix) * 3 + d];
        }
        for (int i = 0; i < 16; ++i)
            for (int j = 0; j < 16; ++j) {
                float s = 0.0f;
                for (int d = 0; d < 3; ++d) {
                    float df = pts[b][i][d] - pts[b][j][d];
                    s += df * df;
                }
                dmat[b][i][j] = sqrtf(fmaxf(s, 0.0f));
            }
        float mind[16];
        unsigned intree = 1u;
        for (int j = 0; j < 16; ++j) mind[j] = dmat[b][0][j];
        for (int step = 0; step < 15; ++step) {
            int jm = 0; float w = 3.4e38f;
            for (int j = 0; j < 16; ++j) {
                bool in = (intree >> j) & 1u;
                if (!in && mind[j] < w) { w = mind[j]; jm = j; }
            }
            intree |= (1u << jm);
            deaths[b][pass][step] = w;
            for (int j = 0; j < 16; ++j) mind[j] = fminf(mind[j], dmat[b][jm][j]);
        }
        for (int i = 1; i < 15; ++i) {        // insertion sort ascending
            float key = deaths[b][pass][i];
            int k = i - 1;
            while (k >= 0 && deaths[b][pass][k] > key) {
                deaths[b][pass][k + 1] = deaths[b][pass][k];
                --k;
            }
            deaths[b][pass][k + 1] = key;
        }
    }
    float s = 0.0f;
    for (int i = 0; i < 15; ++i) {
        float d = deaths[b][0][i] - deaths[b][1][i];
        s += d * d;
    }
    ws_wd[b] = sqrtf(s);                      // (sum |.|^2)^(1/2), Q = 2
}

// ---------------------------------------------------------------------------
// Kernel D: deterministic fixed-order reductions -> 3 scalar outputs.
// ---------------------------------------------------------------------------
__global__ __launch_bounds__(256) void finalize_kernel(
    const float* __restrict__ ws_min, const float* __restrict__ ws_rep,
    const float* __restrict__ ws_wd, const float* __restrict__ pcd_radius,
    float* __restrict__ out)
{
    __shared__ float red[256];
    __shared__ float emd_s;
    const int t = threadIdx.x;
    if (t == 0) emd_s = 0.0f;
    __syncthreads();
    for (int b = 0; b < NB; ++b) {
        float s = 0.0f;
        for (int i = t; i < NPTS; i += 256) s += ws_min[(size_t)b * NPTS + i];
        red[t] = s; __syncthreads();
        for (int k = 128; k > 0; k >>= 1) {
            if (t < k) red[t] += red[t + k];
            __syncthreads();
        }
        if (t == 0) emd_s += (red[0] / (float)(NPTS * 3)) / pcd_radius[b];
        __syncthreads();
    }
    float s = 0.0f;
    for (int i = t; i < NB * NPTS; i += 256) s += ws_rep[i];
    red[t] = s; __syncthreads();
    for (int k = 128; k > 0; k >>= 1) {
        if (t < k) red[t] += red[t + k];
        __syncthreads();
    }
    if (t == 0) {
        out[0] = 100.0f * emd_s / (float)NB;            // emd * 100
        out[1] = red[0] / (float)(NB * NPTS * 4);       // rep * ALPHA (=1)
        float tda = 0.0f;
        for (int b = 0; b < NB; ++b) tda += ws_wd[b];
        out[2] = tda / (float)NB;                       // tda * BETA (=1)
    }
}

// ---------------------------------------------------------------------------
extern "C" void kernel_launch(void* const* d_in, const int* in_sizes, int n_in,
                              void* d_out, int out_size, void* d_ws, size_t ws_size,
                              hipStream_t stream)
{
    (void)in_sizes; (void)n_in; (void)out_size; (void)ws_size;
    const float* pred = (const float*)d_in[0];
    const float* gt   = (const float*)d_in[1];
    const float* pcd  = (const float*)d_in[2];
    float* out = (float*)d_out;

    float* ws_min = (float*)d_ws;                      // NB*NPTS floats
    float* ws_rep = ws_min + (size_t)NB * NPTS;        // NB*NPTS floats
    float* ws_wd  = ws_rep + (size_t)NB * NPTS;        // NB floats

    dim3 grid(NPTS / 128, NB);
    emd_min_kernel<<<grid, 256, 0, stream>>>(pred, gt, ws_min);
    rep_knn_kernel<<<grid, 256, 0, stream>>>(pred, ws_rep);
    tda_kernel<<<1, 32, 0, stream>>>(pred, gt, ws_wd);
    finalize_kernel<<<1, 256, 0, stream>>>(ws_min, ws_rep, ws_wd, pcd, out);
}